// Tacotron2Loss_17360257810661
// MI455X (gfx1250) — compile-verified
//
#include <hip/hip_runtime.h>
#include <hip/hip_bf16.h>

typedef __attribute__((ext_vector_type(2))) float v2f;
typedef __attribute__((ext_vector_type(4))) float v4f;
typedef __attribute__((ext_vector_type(8))) float v8f;

// workspace layout (float offsets)
#define WS_MEL   0      // 1250 partials
#define WS_ALIGN 2048   // 782 partials
#define WS_GATE  3072   // 250 partials
#define WS_SPK   3584   // 64 row losses
#define WS_AUG   3712   // 64 row losses

__device__ __forceinline__ float waveSum(float v) {
#pragma unroll
  for (int o = 16; o > 0; o >>= 1) v += __shfl_xor(v, o, 32);
  return v;
}

// full-block sum; result valid in thread 0. Safe to call repeatedly.
__device__ __forceinline__ float blockSum(float v) {
  __shared__ float red[32];
  const int lane = threadIdx.x & 31, wid = threadIdx.x >> 5;
  v = waveSum(v);
  if (lane == 0) red[wid] = v;
  __syncthreads();
  const int nw = (blockDim.x + 31) >> 5;
  float r = (threadIdx.x < nw) ? red[threadIdx.x] : 0.f;
  if (wid == 0) r = waveSum(r);
  __syncthreads();
  return r;
}

// ---------------------------------------------------------------------------
// Kernel 1: fused mel MSE (mel_out & postnet vs target), WMMA-f32 accumulator.
// grid = 1250 blocks x 256 threads; each thread: 4 iters x float4 -> exactly
// 5,120,000 elements, no tail => EXEC all-ones for every v_wmma.
// ---------------------------------------------------------------------------
__global__ void __launch_bounds__(256)
mel_mse_wmma(const v4f* __restrict__ mo, const v4f* __restrict__ po,
             const v4f* __restrict__ tg, float* __restrict__ partial) {
  const int G = 320000;  // 1250 * 256 threads (in float4 units)
  const int i = blockIdx.x * 256 + threadIdx.x;

  v8f c = {};                 // persistent f32 accumulator (C/D matrix)
  v2f ones; ones[0] = 1.f; ones[1] = 1.f;   // B = all-ones 4x16

#pragma unroll
  for (int k = 0; k < 4; ++k) {
    const int idx = i + k * G;
    v4f a = __builtin_nontemporal_load(&mo[idx]);
    v4f b = __builtin_nontemporal_load(&po[idx]);
    v4f t = __builtin_nontemporal_load(&tg[idx]);
    v4f d1 = a - t;
    v4f d2 = b - t;
    v4f s  = d1 * d1 + d2 * d2;             // per-elem squared error (both streams)
    v2f av; av[0] = s[0] + s[1]; av[1] = s[2] + s[3];
    // D = A x ones + C : every D[m,n] accumulates the rowsum of A
    c = __builtin_amdgcn_wmma_f32_16x16x4_f32(false, av, false, ones,
                                              (short)0, c, false, false);
  }

  // rowsums: lane 0 carries M=0..7 (c[0..7]), lane 16 carries M=8..15
  float s = c[0] + c[1] + c[2] + c[3] + c[4] + c[5] + c[6] + c[7];
  float wave = __shfl(s, 0, 32) + __shfl(s, 16, 32);

  __shared__ float red[8];
  const int lane = threadIdx.x & 31, wid = threadIdx.x >> 5;
  if (lane == 0) red[wid] = wave;
  __syncthreads();
  if (threadIdx.x == 0) {
    float tot = 0.f;
#pragma unroll
    for (int w = 0; w < 8; ++w) tot += red[w];
    partial[blockIdx.x] = tot;
  }
}

// ---------------------------------------------------------------------------
// Kernel 2: alignment penalty. One thread per (n,t); g computed once, then
// 64 coalesced batch-strided NT loads (stride = NDEC*TENC floats).
// ---------------------------------------------------------------------------
__global__ void __launch_bounds__(256)
align_kernel(const float* __restrict__ al, float* __restrict__ partial) {
  const int p = blockIdx.x * 256 + threadIdx.x;   // (n,t) pair, 200,000 total
  float contrib = 0.f;
  if (p < 200000) {
    const int n = p / 200;
    const int t = p - n * 200;
    const float d = (float)n * (1.f / 1000.f) - (float)t * (1.f / 200.f);
    const float g = 1.f - expf(-2.5f * d * d);    // 1/K_ALIGN = 2.5
    const float* base = al + p;
    float s = 0.f;
#pragma unroll 8
    for (int b = 0; b < 64; ++b)
      s += __builtin_nontemporal_load(base + b * 200000);
    contrib = g * s;
  }
  const float tot = blockSum(contrib);
  if (threadIdx.x == 0) partial[blockIdx.x] = tot;
}

// ---------------------------------------------------------------------------
// Kernel 3: gate BCE-with-logits, mean. 64,000 elements exactly.
// ---------------------------------------------------------------------------
__global__ void __launch_bounds__(256)
gate_kernel(const float* __restrict__ x, const float* __restrict__ y,
            float* __restrict__ partial) {
  const int i = blockIdx.x * 256 + threadIdx.x;   // 250 * 256 = 64000 exact
  const float xv = __builtin_nontemporal_load(x + i);
  const float yv = __builtin_nontemporal_load(y + i);
  // stable softplus
  const float sp = fmaxf(xv, 0.f) + log1pf(expf(-fabsf(xv)));
  const float tot = blockSum(sp - xv * yv);
  if (threadIdx.x == 0) partial[blockIdx.x] = tot;
}

// ---------------------------------------------------------------------------
// Kernel 4: per-row cross-entropy vs argmax(target). One wave per row.
// ---------------------------------------------------------------------------
__global__ void __launch_bounds__(32)
ce_rows(const float* __restrict__ logits, const float* __restrict__ target,
        int C, float* __restrict__ out_rows) {
  const int row = blockIdx.x;
  const int lane = threadIdx.x;
  const float* lrow = logits + row * C;
  const float* trow = target + row * C;

  float bestv = -INFINITY; int besti = 0x7fffffff;
  float maxl = -INFINITY;
  for (int cI = lane; cI < C; cI += 32) {
    const float tv = trow[cI];
    if (tv > bestv || (tv == bestv && cI < besti)) { bestv = tv; besti = cI; }
    maxl = fmaxf(maxl, lrow[cI]);
  }
#pragma unroll
  for (int o = 16; o > 0; o >>= 1) {
    const float ov = __shfl_xor(bestv, o, 32);
    const int   oi = __shfl_xor(besti, o, 32);
    if (ov > bestv || (ov == bestv && oi < besti)) { bestv = ov; besti = oi; }
    maxl = fmaxf(maxl, __shfl_xor(maxl, o, 32));
  }
  float sumexp = 0.f;
  for (int cI = lane; cI < C; cI += 32) sumexp += expf(lrow[cI] - maxl);
#pragma unroll
  for (int o = 16; o > 0; o >>= 1) sumexp += __shfl_xor(sumexp, o, 32);

  if (lane == 0)
    out_rows[row] = -(lrow[besti] - maxl - logf(sumexp));
}

// ---------------------------------------------------------------------------
// Kernel 5: combine partials -> 5 output scalars.
// ---------------------------------------------------------------------------
__global__ void __launch_bounds__(256)
finalize_kernel(const float* __restrict__ ws, float* __restrict__ out) {
  float s0 = 0, s1 = 0, s2 = 0, s3 = 0, s4 = 0;
  for (int i = threadIdx.x; i < 1250; i += 256) s0 += ws[WS_MEL + i];
  for (int i = threadIdx.x; i < 782;  i += 256) s1 += ws[WS_ALIGN + i];
  for (int i = threadIdx.x; i < 250;  i += 256) s2 += ws[WS_GATE + i];
  for (int i = threadIdx.x; i < 64;   i += 256) s3 += ws[WS_SPK + i];
  for (int i = threadIdx.x; i < 64;   i += 256) s4 += ws[WS_AUG + i];
  s0 = blockSum(s0); s1 = blockSum(s1); s2 = blockSum(s2);
  s3 = blockSum(s3); s4 = blockSum(s4);
  if (threadIdx.x == 0) {
    const float mel  = s0 * (1.f / 5120000.f);   // both MSE means share the count
    const float gate = s2 * (1.f / 64000.f);
    const float spk  = s3 * (1.f / 64.f);
    const float aug  = s4 * (1.f / 64.f);
    const float alo  = fabsf(s1) * 0.0005f;      // LAMB
    const float tmp  = mel + gate;
    out[0] = 10.f * tmp + 0.1f * (spk + aug) + alo;
    out[1] = tmp;
    out[2] = spk;
    out[3] = aug;
    out[4] = alo;
  }
}

// ---------------------------------------------------------------------------
extern "C" void kernel_launch(void* const* d_in, const int* in_sizes, int n_in,
                              void* d_out, int out_size, void* d_ws, size_t ws_size,
                              hipStream_t stream) {
  const float* mel_out        = (const float*)d_in[0];
  const float* mel_post       = (const float*)d_in[1];
  const float* gate_out       = (const float*)d_in[2];
  const float* alignment     = (const float*)d_in[3];
  const float* speaker_out    = (const float*)d_in[4];
  const float* augment_out    = (const float*)d_in[5];
  const float* mel_target     = (const float*)d_in[6];
  const float* gate_target    = (const float*)d_in[7];
  const float* speaker_target = (const float*)d_in[8];
  const float* aug_target     = (const float*)d_in[9];
  float* ws  = (float*)d_ws;
  float* out = (float*)d_out;

  mel_mse_wmma<<<1250, 256, 0, stream>>>((const v4f*)mel_out, (const v4f*)mel_post,
                                         (const v4f*)mel_target, ws + WS_MEL);
  align_kernel<<<782, 256, 0, stream>>>(alignment, ws + WS_ALIGN);
  gate_kernel<<<250, 256, 0, stream>>>(gate_out, gate_target, ws + WS_GATE);
  ce_rows<<<64, 32, 0, stream>>>(speaker_out, speaker_target, 100, ws + WS_SPK);
  ce_rows<<<64, 32, 0, stream>>>(augment_out, aug_target, 10, ws + WS_AUG);
  finalize_kernel<<<1, 256, 0, stream>>>(ws, out);
}